// GAT_13786845020362
// MI455X (gfx1250) — compile-verified
//
#include <hip/hip_runtime.h>
#include <hip/hip_bf16.h>
#include <math.h>

// ---------------- problem constants (match reference) ----------------
#define N_NODES 20000
#define N_EDGES 160000
#define NFEAT   128
#define H1      4
#define C1      32
#define D1      128      // H1*C1
#define H2      4
#define C2      128
#define D2      512      // H2*C2
#define NCLASS  64
#define NEG_SLOPE 0.2f

typedef float v2f __attribute__((ext_vector_type(2)));
typedef float v8f __attribute__((ext_vector_type(8)));

// ---------------- fp32 WMMA GEMM: C[M,N] = A[M,K] @ W[K,N] + bias, opt. ReLU
// One wave computes a 16x64 output tile: 4 accumulators share one A fragment
// per K-step. All 4 B fragments are loaded into distinct registers BEFORE the
// 4 WMMAs so the compiler can clause the loads, wait once, and issue the
// v_wmma ops back-to-back (overlapping WMMA latency with next-step loads).
// A-frag layout per ISA 7.12.2 (32-bit A 16x4): lanes 0-15 hold K={0,1},
// lanes 16-31 hold K={2,3}. B symmetric (4x16, row striped across lanes).
// Requires: M % 16 == 0, N % 64 == 0, K % 4 == 0 (true for all call sites).
__global__ void wmma_gemm_bias(const float* __restrict__ A,
                               const float* __restrict__ W,
                               const float* __restrict__ bias,
                               float* __restrict__ C,
                               int M, int K, int N, int do_relu) {
    const int tm   = blockIdx.x * 16;
    const int tn0  = blockIdx.y * 64;
    const int lane = threadIdx.x & 31;
    const int half = lane >> 4;      // 0: lanes 0-15, 1: lanes 16-31
    const int l    = lane & 15;

    v8f acc[4];
    #pragma unroll
    for (int t = 0; t < 4; ++t) acc[t] = v8f{};

    // A fragment base: row tm+l, K offset (2*half)
    const float* aRow = A + (size_t)(tm + l) * K + 2 * half;
    // B fragment base: row (2*half), col tn0 + l; advances 4 rows per K-step
    const float* bp    = W + (size_t)(2 * half) * N + (tn0 + l);
    const size_t bstep = (size_t)4 * N;

    for (int k = 0; k < K; k += 4) {
        v2f a;
        a.x = aRow[k];
        a.y = aRow[k + 1];

        // gather all 4 B fragments first (distinct regs -> one load clause)
        float b0[4], b1[4];
        #pragma unroll
        for (int t = 0; t < 4; ++t) {
            b0[t] = bp[16 * t];        // row k+2*half
            b1[t] = bp[N + 16 * t];    // row k+2*half+1
        }

        #pragma unroll
        for (int t = 0; t < 4; ++t) {
            v2f b;
            b.x = b0[t];
            b.y = b1[t];
            acc[t] = __builtin_amdgcn_wmma_f32_16x16x4_f32(
                /*neg_a=*/false, a, /*neg_b=*/false, b,
                /*c_mod=*/(short)0, acc[t], /*reuse_a=*/false, /*reuse_b=*/false);
        }
        bp += bstep;
    }

    // C/D layout: VGPR r -> row (r + 8*half), col = lane&15
    #pragma unroll
    for (int t = 0; t < 4; ++t) {
        const int tn = tn0 + 16 * t;
        const float bv = bias[tn + l];
        #pragma unroll
        for (int r = 0; r < 8; ++r) {
            int row = tm + r + half * 8;
            float v = acc[t][r] + bv;
            if (do_relu) v = fmaxf(v, 0.f);
            C[(size_t)row * N + (tn + l)] = v;
        }
    }
}

// ---------------- helpers ----------------
__global__ void fill_f32(float* __restrict__ p, float v, size_t n) {
    size_t i = (size_t)blockIdx.x * blockDim.x + threadIdx.x;
    if (i < n) p[i] = v;
}

__device__ __forceinline__ void atomicMaxFloat(float* addr, float val) {
    // IEEE trick: non-negative floats ordered like ints; negative reversed in uint.
    if (val >= 0.f) {
        atomicMax((int*)addr, __float_as_int(val));
    } else {
        atomicMin((unsigned int*)addr, __float_as_uint(val));
    }
}

// ---------------- per-(edge,head) attention score + running max ----------------
template <int H, int C>
__global__ void edge_score_kernel(const float* __restrict__ xl,
                                  const float* __restrict__ xr,
                                  const int* __restrict__ src,
                                  const int* __restrict__ dst,
                                  const float* __restrict__ att,
                                  float* __restrict__ score,
                                  float* __restrict__ smax,
                                  int E) {
    int wid  = (int)((blockIdx.x * (size_t)blockDim.x + threadIdx.x) >> 5);
    int lane = threadIdx.x & 31;
    if (wid >= E * H) return;
    int e = wid / H, h = wid % H;
    int s = src[e], d = dst[e];

    const float* pl = xl + (size_t)s * (H * C) + h * C;
    const float* pr = xr + (size_t)d * (H * C) + h * C;
    const float* pa = att + h * C;

    float sc = 0.f;
    for (int c = lane; c < C; c += 32) {
        float v = pl[c] + pr[c];
        v = (v >= 0.f) ? v : NEG_SLOPE * v;   // LeakyReLU(0.2)
        sc += pa[c] * v;
    }
    for (int off = 16; off > 0; off >>= 1) sc += __shfl_down(sc, off, 32);

    if (lane == 0) {
        score[(size_t)e * H + h] = sc;
        atomicMaxFloat(&smax[(size_t)d * H + h], sc);
    }
}

// ---------------- exp(score - max[dst]) and denominator ----------------
template <int H>
__global__ void edge_exp_kernel(float* __restrict__ score,
                                const float* __restrict__ smax,
                                const int* __restrict__ dst,
                                float* __restrict__ denom,
                                int E) {
    int i = (int)(blockIdx.x * (size_t)blockDim.x + threadIdx.x);
    if (i >= E * H) return;
    int e = i / H, h = i % H;
    int d = dst[e];
    float a = expf(score[i] - smax[(size_t)d * H + h]);
    score[i] = a;
    atomicAdd(&denom[(size_t)d * H + h], a);
}

// ---------------- weighted aggregation: out[dst] += a/denom * xl[src] ------
template <int H, int C>
__global__ void edge_aggregate_kernel(const float* __restrict__ xl,
                                      const float* __restrict__ score,
                                      const float* __restrict__ denom,
                                      const int* __restrict__ src,
                                      const int* __restrict__ dst,
                                      float* __restrict__ out,
                                      int E) {
    int wid  = (int)((blockIdx.x * (size_t)blockDim.x + threadIdx.x) >> 5);
    int lane = threadIdx.x & 31;
    if (wid >= E * H) return;
    int e = wid / H, h = wid % H;
    int s = src[e], d = dst[e];

    float den  = denom[(size_t)d * H + h];
    float coef = score[(size_t)e * H + h] / ((den == 0.f) ? 1.f : den);

    const float* pl = xl + (size_t)s * (H * C) + h * C;
    float*       po = out + (size_t)d * (H * C) + h * C;
    for (int c = lane; c < C; c += 32) {
        atomicAdd(&po[c], coef * pl[c]);
    }
}

// ---------------- bias + ReLU, in place ----------------
__global__ void bias_relu_kernel(float* __restrict__ io,
                                 const float* __restrict__ bias,
                                 int M, int N) {
    size_t i = (size_t)blockIdx.x * blockDim.x + threadIdx.x;
    if (i >= (size_t)M * N) return;
    float v = io[i] + bias[i % N];
    io[i] = fmaxf(v, 0.f);
}

// =====================================================================
extern "C" void kernel_launch(void* const* d_in, const int* in_sizes, int n_in,
                              void* d_out, int out_size, void* d_ws, size_t ws_size,
                              hipStream_t stream) {
    const float* x     = (const float*)d_in[0];
    const int*   ei    = (const int*)  d_in[1];
    const int*   src   = ei;              // edge_index[0]
    const int*   dst   = ei + N_EDGES;    // edge_index[1]
    // d_in[2] = edge_attr (unused by reference)
    const float* Wl1   = (const float*)d_in[3];
    const float* bl1   = (const float*)d_in[4];
    const float* Wr1   = (const float*)d_in[5];
    const float* br1   = (const float*)d_in[6];
    const float* att1  = (const float*)d_in[7];
    const float* bias1 = (const float*)d_in[8];
    const float* Wl2   = (const float*)d_in[9];
    const float* bl2   = (const float*)d_in[10];
    const float* Wr2   = (const float*)d_in[11];
    const float* br2   = (const float*)d_in[12];
    const float* att2  = (const float*)d_in[13];
    const float* bias2 = (const float*)d_in[14];
    const float* Wlin  = (const float*)d_in[15];
    const float* blin  = (const float*)d_in[16];
    float* out = (float*)d_out;

    // ---- workspace layout (floats) ----
    float* ws    = (float*)d_ws;
    float* xl    = ws;                                 // [N, 512] (layer1 uses 128 cols)
    float* xr    = xl    + (size_t)N_NODES * D2;       // [N, 512]
    float* h1    = xr    + (size_t)N_NODES * D2;       // [N, 128]
    float* h2    = h1    + (size_t)N_NODES * D1;       // [N, 512]
    float* score = h2    + (size_t)N_NODES * D2;       // [E, 4]
    float* smax  = score + (size_t)N_EDGES * H1;       // [N, 4]
    float* denom = smax  + (size_t)N_NODES * H1;       // [N, 4]

    const float NEG_INF = -__builtin_huge_valf();

    const int waves1   = N_EDGES * H1;                     // 640k (edge,head) waves
    const int wblocks  = (waves1 * 32 + 255) / 256;        // wave-per-(e,h) grids
    const int tblocks  = (waves1 + 255) / 256;             // thread-per-(e,h) grids

    // ================= Layer 1 =================
    {
        dim3 g(N_NODES / 16, D1 / 64);
        wmma_gemm_bias<<<g, 32, 0, stream>>>(x, Wl1, bl1, xl, N_NODES, NFEAT, D1, 0);
        wmma_gemm_bias<<<g, 32, 0, stream>>>(x, Wr1, br1, xr, N_NODES, NFEAT, D1, 0);
    }
    fill_f32<<<(N_NODES * H1 + 255) / 256, 256, 0, stream>>>(smax, NEG_INF, (size_t)N_NODES * H1);
    fill_f32<<<(N_NODES * H1 + 255) / 256, 256, 0, stream>>>(denom, 0.f, (size_t)N_NODES * H1);
    fill_f32<<<((size_t)N_NODES * D1 + 255) / 256, 256, 0, stream>>>(h1, 0.f, (size_t)N_NODES * D1);

    edge_score_kernel<H1, C1><<<wblocks, 256, 0, stream>>>(xl, xr, src, dst, att1, score, smax, N_EDGES);
    edge_exp_kernel<H1><<<tblocks, 256, 0, stream>>>(score, smax, dst, denom, N_EDGES);
    edge_aggregate_kernel<H1, C1><<<wblocks, 256, 0, stream>>>(xl, score, denom, src, dst, h1, N_EDGES);
    bias_relu_kernel<<<((size_t)N_NODES * D1 + 255) / 256, 256, 0, stream>>>(h1, bias1, N_NODES, D1);

    // ================= Layer 2 =================
    {
        dim3 g(N_NODES / 16, D2 / 64);
        wmma_gemm_bias<<<g, 32, 0, stream>>>(h1, Wl2, bl2, xl, N_NODES, D1, D2, 0);
        wmma_gemm_bias<<<g, 32, 0, stream>>>(h1, Wr2, br2, xr, N_NODES, D1, D2, 0);
    }
    fill_f32<<<(N_NODES * H2 + 255) / 256, 256, 0, stream>>>(smax, NEG_INF, (size_t)N_NODES * H2);
    fill_f32<<<(N_NODES * H2 + 255) / 256, 256, 0, stream>>>(denom, 0.f, (size_t)N_NODES * H2);
    fill_f32<<<((size_t)N_NODES * D2 + 255) / 256, 256, 0, stream>>>(h2, 0.f, (size_t)N_NODES * D2);

    edge_score_kernel<H2, C2><<<wblocks, 256, 0, stream>>>(xl, xr, src, dst, att2, score, smax, N_EDGES);
    edge_exp_kernel<H2><<<tblocks, 256, 0, stream>>>(score, smax, dst, denom, N_EDGES);
    edge_aggregate_kernel<H2, C2><<<wblocks, 256, 0, stream>>>(xl, score, denom, src, dst, h2, N_EDGES);
    bias_relu_kernel<<<((size_t)N_NODES * D2 + 255) / 256, 256, 0, stream>>>(h2, bias2, N_NODES, D2);

    // ================= Output linear =================
    {
        dim3 g(N_NODES / 16, NCLASS / 64);
        wmma_gemm_bias<<<g, 32, 0, stream>>>(h2, Wlin, blin, out, N_NODES, D2, NCLASS, 0);
    }
}